// WRF_90159953477734
// MI455X (gfx1250) — compile-verified
//
#include <hip/hip_runtime.h>
#include <math.h>

// ---------------- problem constants ----------------
#define NZv 48
#define NYv 256
#define NXv 256
#define NZW 47          // W / O / phi1 vertical levels
#define NXF 255         // U faces in x
#define NYF 255         // V faces in y

constexpr float cDX = 500.f, cDY = 500.f, cDZ = 1.f / 48.f;
constexpr float cG = 9.8f, cGAM = 1.4f, cPREF = 101325.f, cRD = 287.f;
constexpr float cPTOP = 25000.f, cTH0 = 300.f;
constexpr float cPIC = cPREF - cPTOP;   // mu0

constexpr long SZC = (long)NZv * NYv * NXv;
constexpr long SZU = (long)NZv * NYv * NXF;
constexpr long SZV = (long)NZv * NYF * NXv;
constexpr long SZW = (long)NZW * NYv * NXv;
constexpr long SZP = (long)NYv * NXv;

// ---------------- index helpers ----------------
__device__ __forceinline__ long IC(int z, int y, int x) { return ((long)z * NYv + y) * NXv + x; }
__device__ __forceinline__ long IW(int z, int y, int x) { return ((long)z * NYv + y) * NXv + x; }
__device__ __forceinline__ long IU(int z, int y, int x) { return ((long)z * NYv + y) * NXF + x; }
__device__ __forceinline__ long IV(int z, int y, int x) { return ((long)z * NYF + y) * NXv + x; }
__device__ __forceinline__ int  wfx(int g) { return (g + NXF) % NXF; }   // periodic face wrap

__device__ __forceinline__ float muS(const float* mu1, int z, int y, int x) {
  return cPIC + mu1[IC(z, y, x)];
}
// bz_(pz_(f)) for f on 47-level grid, evaluated at cell z (0..47)
__device__ __forceinline__ float bzpzW(const float* f, int z, int y, int x) {
  float lo = (z > 0) ? f[IW(z - 1, y, x)] : 0.f;
  float hi = (z < NZW) ? f[IW(z, y, x)] : 0.f;
  return 0.5f * (lo + hi);
}
// dz_(pz_(f)) at cell z
__device__ __forceinline__ float dzpzW(const float* f, int z, int y, int x) {
  float lo = (z > 0) ? f[IW(z - 1, y, x)] : 0.f;
  float hi = (z < NZW) ? f[IW(z, y, x)] : 0.f;
  return (hi - lo) / cDZ;
}
// contravariant velocities
__device__ __forceinline__ float ufc(const float* U, const float* mu1, int z, int y, int g) {
  return U[IU(z, y, g)] / (0.5f * (muS(mu1, z, y, g) + muS(mu1, z, y, g + 1)));
}
__device__ __forceinline__ float vfc(const float* V, const float* mu1, int z, int g, int x) {
  return V[IV(z, g, x)] / (0.5f * (muS(mu1, z, g, x) + muS(mu1, z, g + 1, x)));
}
// bx_(px_(u)) at cell x
__device__ __forceinline__ float cxu(const float* U, const float* mu1, int z, int y, int x) {
  return 0.5f * (ufc(U, mu1, z, y, wfx(x - 1)) + ufc(U, mu1, z, y, wfx(x)));
}
// by_(py_(v)) at cell y
__device__ __forceinline__ float cyv(const float* V, const float* mu1, int z, int y, int x) {
  float a = (y > 0) ? vfc(V, mu1, z, y - 1, x) : 0.f;
  float b = (y < NYF) ? vfc(V, mu1, z, y, x) : 0.f;
  return 0.5f * (a + b);
}

// ---------------- base-state profiles ----------------
// prof[0..47]=p0(z), prof[96..144]=phi0f(k=0..48), prof[145]=ztop
__global__ void kprof(float* prof) {
  if (threadIdx.x == 0 && blockIdx.x == 0) {
    float p0[NZv], a0[NZv];
    for (int z = 0; z < NZv; z++) {
      p0[z] = cPTOP + cPIC * ((z + 0.5f) / 48.f);
      a0[z] = cRD * cTH0 / cPREF * powf(p0[z] / cPREF, -1.f / cGAM);
      prof[z] = p0[z];
      prof[48 + z] = a0[z];
    }
    float base = a0[NZv - 1] * cPIC;
    prof[96 + 48] = base;
    float run = base;
    for (int z = NZv - 1; z >= 0; z--) { run += a0[z] * cPIC * cDZ; prof[96 + z] = run; }
    prof[145] = prof[96] / cG;
  }
}

// ---------------- fixed (step_fn) derived fields: from ORIGINAL inputs ----------------
__global__ void kfixed(const float* Th, const float* phi1, const float* mu1, const float* prof,
                       float* ALF, float* PF, float* CcF, float* GPTF, float* THF, float* DZPHI) {
  long i = (long)blockIdx.x * blockDim.x + threadIdx.x;
  if (i >= SZC) return;
  int x = (int)(i % NXv), y = (int)((i / NXv) % NYv), z = (int)(i / ((long)NXv * NYv));
  float mu = cPIC + mu1[i];
  float plo = prof[96 + z] + ((z >= 1) ? phi1[IW(z - 1, y, x)] : 0.f);
  float phi_ = prof[96 + z + 1] + ((z < NZW) ? phi1[IW(z, y, x)] : 0.f);
  float alpha = -(phi_ - plo) / cDZ / mu;
  float th = Th[i] / mu;
  float p = cPREF * powf(cRD * th / (cPREF * alpha), cGAM);
  ALF[i] = alpha; PF[i] = p;
  CcF[i] = cGAM * p / (mu * alpha);
  GPTF[i] = cGAM * p / Th[i];
  THF[i] = th;
  if (z < NZW) {
    float a = prof[96 + z] + ((z >= 1) ? phi1[IW(z - 1, y, x)] : 0.f);
    float b = prof[96 + z + 2] + ((z + 2 <= NZW) ? phi1[IW(z + 1, y, x)] : 0.f);
    DZPHI[IW(z, y, x)] = 0.5f * (b - a) / cDZ;
  }
}

__global__ void kdpx(const float* phi1, float* DPX) {
  long i = (long)blockIdx.x * blockDim.x + threadIdx.x;
  if (i >= SZU) return;
  int xf = (int)(i % NXF), y = (int)((i / NXF) % NYv), z = (int)(i / ((long)NXF * NYv));
  DPX[i] = (bzpzW(phi1, z, y, xf + 1) - bzpzW(phi1, z, y, xf)) / cDX;
}
__global__ void kdpy(const float* phi1, float* DPY) {
  long i = (long)blockIdx.x * blockDim.x + threadIdx.x;
  if (i >= SZV) return;
  int x = (int)(i % NXv), yf = (int)((i / NXv) % NYF), z = (int)(i / ((long)NXv * NYF));
  DPY[i] = (bzpzW(phi1, z, yf + 1, x) - bzpzW(phi1, z, yf, x)) / cDY;
}

// ---------------- per-stage derived fields for RHS(state) ----------------
__global__ void kderive(const float* Th, const float* phi1, const float* mu1, const float* prof,
                        float* ALR, float* P1R, float* THR) {
  long i = (long)blockIdx.x * blockDim.x + threadIdx.x;
  if (i >= SZC) return;
  int x = (int)(i % NXv), y = (int)((i / NXv) % NYv), z = (int)(i / ((long)NXv * NYv));
  float mu = cPIC + mu1[i];
  float plo = prof[96 + z] + ((z >= 1) ? phi1[IW(z - 1, y, x)] : 0.f);
  float phi_ = prof[96 + z + 1] + ((z < NZW) ? phi1[IW(z, y, x)] : 0.f);
  float alpha = -(phi_ - plo) / cDZ / mu;
  float th = Th[i] / mu;
  float p = cPREF * powf(cRD * th / (cPREF * alpha), cGAM);
  ALR[i] = alpha; P1R[i] = p - prof[z]; THR[i] = th;
}

// ---------------- RHS kernels ----------------
__global__ void krU(const float* U, const float* V, const float* O,
                    const float* phi1, const float* mu1,
                    const float* ALR, const float* P1R, float* RU) {
  long i = (long)blockIdx.x * blockDim.x + threadIdx.x;
  if (i >= SZU) return;
  int xf = (int)(i % NXF), y = (int)((i / NXF) % NYv), z = (int)(i / ((long)NXF * NYv));
  float bxmu = 0.5f * (muS(mu1, z, y, xf) + muS(mu1, z, y, xf + 1));
  float dphx = (bzpzW(phi1, z, y, xf + 1) - bzpzW(phi1, z, y, xf)) / cDX;
  float bxal = 0.5f * (ALR[IC(z, y, xf)] + ALR[IC(z, y, xf + 1)]);
  float dxp1 = (P1R[IC(z, y, xf + 1)] - P1R[IC(z, y, xf)]) / cDX;
  auto qbv = [&](int k) -> float {
    if (k == 0) return 0.f;
    int j = (k <= NZW) ? (k - 1) : (NZW - 1);
    return 0.25f * (P1R[IC(j, y, xf)] + P1R[IC(j + 1, y, xf)] +
                    P1R[IC(j, y, xf + 1)] + P1R[IC(j + 1, y, xf + 1)]);
  };
  float dzP = (qbv(z + 1) - qbv(z)) / cDZ;
  float bxmu1 = 0.5f * (mu1[IC(z, y, xf)] + mu1[IC(z, y, xf + 1)]);
  auto Ff = [&](int g) -> float { float uu = ufc(U, mu1, z, y, g); return U[IU(z, y, g)] * uu; };
  float advx = 0.5f * (Ff(wfx(xf + 1)) - Ff(wfx(xf - 1))) / cDX;
  auto Gv = [&](int yf, int xx) -> float {
    return V[IV(z, yf, xx)] * 0.5f * (cxu(U, mu1, z, yf, xx) + cxu(U, mu1, z, yf + 1, xx));
  };
  auto Hrow = [&](int j) -> float {
    if (j < 1 || j > NYF) return 0.f;
    return 0.5f * (Gv(j - 1, xf) + Gv(j - 1, xf + 1));
  };
  float advy = (Hrow(y + 1) - Hrow(y)) / cDY;
  auto Kz = [&](int zf, int xx) -> float {
    return O[IW(zf, y, xx)] * 0.5f * (cxu(U, mu1, zf, y, xx) + cxu(U, mu1, zf + 1, y, xx));
  };
  auto qK = [&](int k) -> float {
    if (k < 1 || k > NZW) return 0.f;
    return 0.5f * (Kz(k - 1, xf) + Kz(k - 1, xf + 1));
  };
  float advz = (qK(z + 1) - qK(z)) / cDZ;
  RU[i] = -bxmu * (dphx + bxal * dxp1) - dphx * (dzP - bxmu1) - advx - advy - advz;
}

__global__ void krV(const float* U, const float* V, const float* O,
                    const float* phi1, const float* mu1,
                    const float* ALR, const float* P1R, float* RV) {
  long i = (long)blockIdx.x * blockDim.x + threadIdx.x;
  if (i >= SZV) return;
  int x = (int)(i % NXv), yf = (int)((i / NXv) % NYF), z = (int)(i / ((long)NXv * NYF));
  float bymu = 0.5f * (muS(mu1, z, yf, x) + muS(mu1, z, yf + 1, x));
  float dphy = (bzpzW(phi1, z, yf + 1, x) - bzpzW(phi1, z, yf, x)) / cDY;
  float byal = 0.5f * (ALR[IC(z, yf, x)] + ALR[IC(z, yf + 1, x)]);
  float dyp1 = (P1R[IC(z, yf + 1, x)] - P1R[IC(z, yf, x)]) / cDY;
  auto qbv = [&](int k) -> float {
    if (k == 0) return 0.f;
    int j = (k <= NZW) ? (k - 1) : (NZW - 1);
    return 0.25f * (P1R[IC(j, yf, x)] + P1R[IC(j + 1, yf, x)] +
                    P1R[IC(j, yf + 1, x)] + P1R[IC(j + 1, yf + 1, x)]);
  };
  float dzP = (qbv(z + 1) - qbv(z)) / cDZ;
  float bymu1 = 0.5f * (mu1[IC(z, yf, x)] + mu1[IC(z, yf + 1, x)]);
  auto Fu = [&](int yy, int g) -> float {
    return U[IU(z, yy, g)] * 0.5f * (cyv(V, mu1, z, yy, g) + cyv(V, mu1, z, yy, g + 1));
  };
  float advx = (0.5f * (Fu(yf, wfx(x)) + Fu(yf + 1, wfx(x))) -
                0.5f * (Fu(yf, wfx(x - 1)) + Fu(yf + 1, wfx(x - 1)))) / cDX;
  auto Fv = [&](int g) -> float { return V[IV(z, g, x)] * vfc(V, mu1, z, g, x); };
  auto byF = [&](int j) -> float {
    float a = (j >= 1) ? Fv(j - 1) : 0.f;
    float b = (j <= NYF - 1) ? Fv(j) : 0.f;
    return 0.5f * (a + b);
  };
  float advy = (byF(yf + 1) - byF(yf)) / cDY;
  auto Kv = [&](int zf, int yy) -> float {
    return O[IW(zf, yy, x)] * 0.5f * (cyv(V, mu1, zf, yy, x) + cyv(V, mu1, zf + 1, yy, x));
  };
  auto qK = [&](int k) -> float {
    if (k < 1 || k > NZW) return 0.f;
    return 0.5f * (Kv(k - 1, yf) + Kv(k - 1, yf + 1));
  };
  float advz = (qK(z + 1) - qK(z)) / cDZ;
  RV[i] = -bymu * (dphy + byal * dyp1) - dphy * (dzP - bymu1) - advx - advy - advz;
}

__global__ void krW(const float* U, const float* V, const float* W, const float* O,
                    const float* mu1, const float* P1R, float* RW) {
  long i = (long)blockIdx.x * blockDim.x + threadIdx.x;
  if (i >= SZW) return;
  int x = (int)(i % NXv), y = (int)((i / NXv) % NYv), zf = (int)(i / ((long)NXv * NYv));
  float dzp1 = (P1R[IC(zf + 1, y, x)] - P1R[IC(zf, y, x)]) / cDZ;
  float bzmu1 = 0.5f * (mu1[IC(zf, y, x)] + mu1[IC(zf + 1, y, x)]);
  auto wc = [&](int zz, int yy, int xx) -> float {
    return W[IW(zz, yy, xx)] / (0.5f * (muS(mu1, zz, yy, xx) + muS(mu1, zz + 1, yy, xx)));
  };
  auto Fx = [&](int g) -> float {
    return 0.5f * (U[IU(zf, y, g)] + U[IU(zf + 1, y, g)]) * 0.5f * (wc(zf, y, g) + wc(zf, y, g + 1));
  };
  float advx = (Fx(wfx(x)) - Fx(wfx(x - 1))) / cDX;
  auto Fy = [&](int g) -> float {
    return 0.5f * (V[IV(zf, g, x)] + V[IV(zf + 1, g, x)]) * 0.5f * (wc(zf, g, x) + wc(zf, g + 1, x));
  };
  float advy = (((y <= NYF - 1) ? Fy(y) : 0.f) - ((y > 0) ? Fy(y - 1) : 0.f)) / cDY;
  auto Ow = [&](int k) -> float { return O[IW(k, y, x)] * wc(k, y, x); };
  float advz = 0.5f * (((zf < NZW - 1) ? Ow(zf + 1) : 0.f) - ((zf > 0) ? Ow(zf - 1) : 0.f)) / cDZ;
  RW[i] = cG * dzp1 - cG * bzmu1 - advx - advy - advz;
}

__global__ void krTh(const float* U, const float* V, const float* O, const float* THR, float* RTh) {
  long i = (long)blockIdx.x * blockDim.x + threadIdx.x;
  if (i >= SZC) return;
  int x = (int)(i % NXv), y = (int)((i / NXv) % NYv), z = (int)(i / ((long)NXv * NYv));
  auto Fx = [&](int g) -> float { return U[IU(z, y, g)] * 0.5f * (THR[IC(z, y, g)] + THR[IC(z, y, g + 1)]); };
  float advx = (Fx(wfx(x)) - Fx(wfx(x - 1))) / cDX;
  auto Fy = [&](int g) -> float { return V[IV(z, g, x)] * 0.5f * (THR[IC(z, g, x)] + THR[IC(z, g + 1, x)]); };
  float advy = (((y < NYF) ? Fy(y) : 0.f) - ((y > 0) ? Fy(y - 1) : 0.f)) / cDY;
  auto Kz = [&](int k) -> float { return O[IW(k, y, x)] * 0.5f * (THR[IC(k, y, x)] + THR[IC(k + 1, y, x)]); };
  float advz = (((z < NZW) ? Kz(z) : 0.f) - ((z > 0) ? Kz(z - 1) : 0.f)) / cDZ;
  RTh[i] = -(advx + advy + advz);
}

__global__ void krPhi(const float* U, const float* V, const float* W, const float* O,
                      const float* phi1, const float* mu1, const float* prof, float* Rphi) {
  long i = (long)blockIdx.x * blockDim.x + threadIdx.x;
  if (i >= SZW) return;
  int x = (int)(i % NXv), y = (int)((i / NXv) % NYv), zf = (int)(i / ((long)NXv * NYv));
  auto Ax = [&](int g) -> float {
    float dxph = (phi1[IW(zf, y, g + 1)] - phi1[IW(zf, y, g)]) / cDX;
    return 0.5f * (U[IU(zf, y, g)] + U[IU(zf + 1, y, g)]) * dxph;
  };
  float tx = 0.5f * (Ax(wfx(x - 1)) + Ax(wfx(x)));
  auto By = [&](int g) -> float {
    float dyph = (phi1[IW(zf, g + 1, x)] - phi1[IW(zf, g, x)]) / cDY;
    return 0.5f * (V[IV(zf, g, x)] + V[IV(zf + 1, g, x)]) * dyph;
  };
  float ty = 0.5f * (((y > 0) ? By(y - 1) : 0.f) + ((y < NYF) ? By(y) : 0.f));
  float pka = prof[96 + zf] + ((zf >= 1) ? phi1[IW(zf - 1, y, x)] : 0.f);
  float pkb = prof[96 + zf + 2] + ((zf + 2 <= NZW) ? phi1[IW(zf + 1, y, x)] : 0.f);
  float dzphi = 0.5f * (pkb - pka) / cDZ;
  float tz = O[IW(zf, y, x)] * dzphi;
  float bzmu = 0.5f * (muS(mu1, zf, y, x) + muS(mu1, zf + 1, y, x));
  Rphi[i] = (-tx - ty - tz + cG * W[IW(zf, y, x)]) / bzmu;
}

__global__ void krMu(const float* U, const float* V, const float* O, float* RMu) {
  long i = (long)blockIdx.x * blockDim.x + threadIdx.x;
  if (i >= SZC) return;
  int x = (int)(i % NXv), y = (int)((i / NXv) % NYv), z = (int)(i / ((long)NXv * NYv));
  float ax = (U[IU(z, y, wfx(x))] - U[IU(z, y, wfx(x - 1))]) / cDX;
  float ay = (((y < NYF) ? V[IV(z, y, x)] : 0.f) - ((y > 0) ? V[IV(z, y - 1, x)] : 0.f)) / cDY;
  float az = (((z < NZW) ? O[IW(z, y, x)] : 0.f) - ((z > 0) ? O[IW(z - 1, y, x)] : 0.f)) / cDZ;
  RMu[i] = -(ax + ay + az);
}

// ---------------- acoustic substep kernels ----------------
__global__ void kp2(const float* Th2, const float* ph2, const float* pi2o,
                    const float* ThIn, const float* mu1in, const float* ALF, const float* PF,
                    float* P2t) {
  long i = (long)blockIdx.x * blockDim.x + threadIdx.x;
  if (i >= SZC) return;
  int x = (int)(i % NXv), y = (int)((i / NXv) % NYv), z = (int)(i / ((long)NXv * NYv));
  float mu = cPIC + mu1in[i];
  float m2 = pi2o[(long)y * NXv + x];
  float a2 = -(dzpzW(ph2, z, y, x) + ALF[i] * m2) / mu;
  P2t[i] = cGAM * PF[i] * (Th2[i] / ThIn[i] - a2 / ALF[i] - m2 / mu);
}

__global__ void kU2(const float* U2o, const float* RU, const float* P2t, const float* ph2,
                    const float* pi2o, const float* mu1in, const float* ALF, const float* DPX,
                    float dtau, float* U2n) {
  long i = (long)blockIdx.x * blockDim.x + threadIdx.x;
  if (i >= SZU) return;
  int xf = (int)(i % NXF), y = (int)((i / NXF) % NYv), z = (int)(i / ((long)NXF * NYv));
  float bxmu = 0.5f * (muS(mu1in, z, y, xf) + muS(mu1in, z, y, xf + 1));
  float bxal = 0.5f * (ALF[IC(z, y, xf)] + ALF[IC(z, y, xf + 1)]);
  float dxp2 = (P2t[IC(z, y, xf + 1)] - P2t[IC(z, y, xf)]) / cDX;
  float dxph2 = (bzpzW(ph2, z, y, xf + 1) - bzpzW(ph2, z, y, xf)) / cDX;
  auto qbv = [&](int k) -> float {
    if (k == 0) return 0.f;
    int j = (k <= NZW) ? (k - 1) : (NZW - 1);
    return 0.25f * (P2t[IC(j, y, xf)] + P2t[IC(j + 1, y, xf)] +
                    P2t[IC(j, y, xf + 1)] + P2t[IC(j + 1, y, xf + 1)]);
  };
  float dzP = (qbv(z + 1) - qbv(z)) / cDZ;
  float bxmu2 = 0.5f * (pi2o[(long)y * NXv + xf] + pi2o[(long)y * NXv + xf + 1]);
  U2n[i] = U2o[i] + dtau * (RU[i] - bxmu * (bxal * dxp2 + dxph2) - DPX[i] * (dzP - bxmu2));
}

__global__ void kV2(const float* V2o, const float* RV, const float* P2t, const float* ph2,
                    const float* pi2o, const float* mu1in, const float* ALF, const float* DPY,
                    float dtau, float* V2n) {
  long i = (long)blockIdx.x * blockDim.x + threadIdx.x;
  if (i >= SZV) return;
  int x = (int)(i % NXv), yf = (int)((i / NXv) % NYF), z = (int)(i / ((long)NXv * NYF));
  float bymu = 0.5f * (muS(mu1in, z, yf, x) + muS(mu1in, z, yf + 1, x));
  float byal = 0.5f * (ALF[IC(z, yf, x)] + ALF[IC(z, yf + 1, x)]);
  float dyp2 = (P2t[IC(z, yf + 1, x)] - P2t[IC(z, yf, x)]) / cDY;
  float dyph2 = (bzpzW(ph2, z, yf + 1, x) - bzpzW(ph2, z, yf, x)) / cDY;
  auto qbv = [&](int k) -> float {
    if (k == 0) return 0.f;
    int j = (k <= NZW) ? (k - 1) : (NZW - 1);
    return 0.25f * (P2t[IC(j, yf, x)] + P2t[IC(j + 1, yf, x)] +
                    P2t[IC(j, yf + 1, x)] + P2t[IC(j + 1, yf + 1, x)]);
  };
  float dzP = (qbv(z + 1) - qbv(z)) / cDZ;
  float bymu2 = 0.5f * (pi2o[(long)yf * NXv + x] + pi2o[(long)(yf + 1) * NXv + x]);
  V2n[i] = V2o[i] + dtau * (RV[i] - bymu * (byal * dyp2 + dyph2) - DPY[i] * (dzP - bymu2));
}

__global__ void kdpi(const float* U2n, const float* V2n, const float* Uin, const float* Vin,
                     const float* pi2o, float dtau, float* dpi, float* pi2n) {
  long t = (long)blockIdx.x * blockDim.x + threadIdx.x;
  if (t >= SZP) return;
  int x = (int)(t % NXv), y = (int)(t / NXv);
  int g1 = wfx(x), g0 = wfx(x - 1);
  float s = 0.f;
  for (int z = 0; z < NZv; z++) {
    float fa = (U2n[IU(z, y, g1)] + Uin[IU(z, y, g1)]) - (U2n[IU(z, y, g0)] + Uin[IU(z, y, g0)]);
    float fb = ((y < NYF) ? (V2n[IV(z, y, x)] + Vin[IV(z, y, x)]) : 0.f) -
               ((y > 0) ? (V2n[IV(z, y - 1, x)] + Vin[IV(z, y - 1, x)]) : 0.f);
    s += fa / cDX + fb / cDY;
    if (z + 1 < NZv) {  // global_prefetch_b8 for the next z-plane
      __builtin_prefetch(&U2n[IU(z + 1, y, g1)], 0, 1);
      __builtin_prefetch(&Uin[IU(z + 1, y, g1)], 0, 1);
    }
  }
  float d = -s * cDZ;
  dpi[t] = d;
  pi2n[t] = pi2o[t] + d * dtau;
}

__global__ void kdv(const float* U2n, const float* V2n, const float* RMu, float* Dv) {
  long i = (long)blockIdx.x * blockDim.x + threadIdx.x;
  if (i >= SZC) return;
  int x = (int)(i % NXv), y = (int)((i / NXv) % NYv), z = (int)(i / ((long)NXv * NYv));
  float ax = (U2n[IU(z, y, wfx(x))] - U2n[IU(z, y, wfx(x - 1))]) / cDX;
  float ay = (((y < NYF) ? V2n[IV(z, y, x)] : 0.f) - ((y > 0) ? V2n[IV(z, y - 1, x)] : 0.f)) / cDY;
  Dv[i] = ax + ay - RMu[i];
}

// ---------------- WMMA reverse-cumsum:  S = T(48x48 upper-tri ones) x Dv(48 x 16cols) ----------------
typedef __attribute__((ext_vector_type(2))) float v2f;
typedef __attribute__((ext_vector_type(8))) float v8f;

#if defined(__AMDGCN__) && __has_builtin(__builtin_amdgcn_wmma_f32_16x16x4_f32)
#define HAVE_WMMA_F32X4 1
#else
#define HAVE_WMMA_F32X4 0
#endif

__global__ void kcum(const float* __restrict__ Dv, const float* __restrict__ dpi,
                     float* __restrict__ O2n) {
  int lane = (int)(threadIdx.x & 31);
  int wv = (int)(threadIdx.x >> 5);
  int y = blockIdx.y;
  int x0 = (blockIdx.x * 4 + wv) * 16;
  int n = lane & 15;
  int half = lane >> 4;
  float dp = dpi[(long)y * NXv + x0 + n];
#if HAVE_WMMA_F32X4
  v8f acc0 = {}, acc1 = {}, acc2 = {};
#pragma unroll
  for (int kb = 0; kb < 12; kb++) {
    int zb = kb * 4 + 2 * half;              // K rows held by this half-wave
    v2f b; b[0] = Dv[IC(zb, y, x0 + n)]; b[1] = Dv[IC(zb + 1, y, x0 + n)];
    int kg0 = zb, kg1 = zb + 1;
    v2f a0, a1, a2;                           // triangular A tiles synthesized in VALU
    a0[0] = (kg0 >= 0 + n) ? 1.f : 0.f;  a0[1] = (kg1 >= 0 + n) ? 1.f : 0.f;
    a1[0] = (kg0 >= 16 + n) ? 1.f : 0.f; a1[1] = (kg1 >= 16 + n) ? 1.f : 0.f;
    a2[0] = (kg0 >= 32 + n) ? 1.f : 0.f; a2[1] = (kg1 >= 32 + n) ? 1.f : 0.f;
    acc0 = __builtin_amdgcn_wmma_f32_16x16x4_f32(false, a0, false, b, (short)0, acc0, false, false);
    acc1 = __builtin_amdgcn_wmma_f32_16x16x4_f32(false, a1, false, b, (short)0, acc1, false, false);
    acc2 = __builtin_amdgcn_wmma_f32_16x16x4_f32(false, a2, false, b, (short)0, acc2, false, false);
  }
#pragma unroll
  for (int r = 0; r < 8; r++) {
    int zg = r + 8 * half;                    // tile 0 rows
    if (zg >= 1)
      O2n[IW(zg - 1, y, x0 + n)] = (float)(NZv - 1 - (zg - 1)) * cDZ * dp + cDZ * acc0[r];
    zg = 16 + r + 8 * half;                   // tile 1
    O2n[IW(zg - 1, y, x0 + n)] = (float)(NZv - 1 - (zg - 1)) * cDZ * dp + cDZ * acc1[r];
    zg = 32 + r + 8 * half;                   // tile 2
    O2n[IW(zg - 1, y, x0 + n)] = (float)(NZv - 1 - (zg - 1)) * cDZ * dp + cDZ * acc2[r];
  }
#else
  if (half == 0) {
    float s = 0.f, Sv[NZv];
    for (int z = NZv - 1; z >= 0; z--) { s += Dv[IC(z, y, x0 + n)]; Sv[z] = s; }
    for (int j = 0; j < NZW; j++)
      O2n[IW(j, y, x0 + n)] = (float)(NZv - 1 - j) * cDZ * dp + cDZ * Sv[j + 1];
  }
#endif
}

__global__ void kth2(const float* Th2o, const float* RTh, const float* U2n, const float* V2n,
                     const float* O2n, const float* THF, float dtau, float* Th2n) {
  long i = (long)blockIdx.x * blockDim.x + threadIdx.x;
  if (i >= SZC) return;
  int x = (int)(i % NXv), y = (int)((i / NXv) % NYv), z = (int)(i / ((long)NXv * NYv));
  auto Fx = [&](int g) -> float { return U2n[IU(z, y, g)] * 0.5f * (THF[IC(z, y, g)] + THF[IC(z, y, g + 1)]); };
  float advx = (Fx(wfx(x)) - Fx(wfx(x - 1))) / cDX;
  auto Fy = [&](int g) -> float { return V2n[IV(z, g, x)] * 0.5f * (THF[IC(z, g, x)] + THF[IC(z, g + 1, x)]); };
  float advy = (((y < NYF) ? Fy(y) : 0.f) - ((y > 0) ? Fy(y - 1) : 0.f)) / cDY;
  auto Kz = [&](int k) -> float { return O2n[IW(k, y, x)] * 0.5f * (THF[IC(k, y, x)] + THF[IC(k + 1, y, x)]); };
  float advz = (((z < NZW) ? Kz(z) : 0.f) - ((z > 0) ? Kz(z - 1) : 0.f)) / cDZ;
  Th2n[i] = Th2o[i] + dtau * (RTh[i] - advx - advy - advz);
}

// phi predictor with x=0:  P = ph2 + dtau*(Rphi - (O2*dzphi - G*W2o/2)/bzmu)
__global__ void kpb(const float* ph2o, const float* Rphi, const float* O2n, const float* DZPHI,
                    const float* W2o, const float* mu1in, float dtau, float* ph2n) {
  long i = (long)blockIdx.x * blockDim.x + threadIdx.x;
  if (i >= SZW) return;
  int x = (int)(i % NXv), y = (int)((i / NXv) % NYv), zf = (int)(i / ((long)NXv * NYv));
  float bzmu = 0.5f * (muS(mu1in, zf, y, x) + muS(mu1in, zf + 1, y, x));
  ph2n[i] = ph2o[i] + dtau * (Rphi[i] - (O2n[i] * DZPHI[i] - cG * W2o[i] * 0.5f) / bzmu);
}

// implicit vertical solve: (I - kap*E(a.)) x = b, Thomas per column, then Rayleigh damping
__global__ void kwsolve(const float* Cc, const float* gpT, const float* Th2n, const float* Th2o,
                        const float* Pb, const float* ph2o, const float* RW, const float* W2o,
                        const float* pi2n, const float* pi2o, const float* mu1in,
                        const float* phi1in, const float* prof, float dtau, float* W2n) {
  long t = (long)blockIdx.x * blockDim.x + threadIdx.x;
  if (t >= SZP) return;
  int x = (int)(t % NXv), y = (int)(t / NXv);
  float kap = dtau * 0.5f * cG / (cDZ * cDZ);
  float piN = pi2n[t], piO = pi2o[t];
  float cp[NZW], dpv[NZW];
  auto aw = [&](int zf) -> float {
    return dtau * cG * 0.5f / (0.5f * (muS(mu1in, zf, y, x) + muS(mu1in, zf + 1, y, x)));
  };
#pragma unroll
  for (int zf = 0; zf < NZW; zf++) {
    float Cz = Cc[IC(zf, y, x)], Czp = Cc[IC(zf + 1, y, x)];
    float az = aw(zf);
    float L = (zf > 0) ? (-kap * Cz * aw(zf - 1)) : 0.f;
    float Du = 1.f + kap * (Cz + Czp) * az;
    float Uu = (zf < NZW - 1) ? (-kap * Czp * aw(zf + 1)) : 0.f;
    float Pm = (zf > 0) ? Pb[IW(zf - 1, y, x)] : 0.f;
    float Pz = Pb[IW(zf, y, x)];
    float Pp = (zf < NZW - 1) ? Pb[IW(zf + 1, y, x)] : 0.f;
    float EP = (Czp * (Pp - Pz) - Cz * (Pz - Pm)) / (cDZ * cDZ);
    float Qm = (zf > 0) ? ph2o[IW(zf - 1, y, x)] : 0.f;
    float Qz = ph2o[IW(zf, y, x)];
    float Qp = (zf < NZW - 1) ? ph2o[IW(zf + 1, y, x)] : 0.f;
    float EO = (Czp * (Qp - Qz) - Cz * (Qz - Qm)) / (cDZ * cDZ);
    float dgn = (gpT[IC(zf + 1, y, x)] * Th2n[IC(zf + 1, y, x)] -
                 gpT[IC(zf, y, x)] * Th2n[IC(zf, y, x)]) / cDZ;
    float dgo = (gpT[IC(zf + 1, y, x)] * Th2o[IC(zf + 1, y, x)] -
                 gpT[IC(zf, y, x)] * Th2o[IC(zf, y, x)]) / cDZ;
    float bb = W2o[IW(zf, y, x)] +
               dtau * (RW[IW(zf, y, x)] + 0.5f * cG * (EP + dgn - piN + EO + dgo - piO));
    float cpm = (zf > 0) ? cp[zf - 1] : 0.f;
    float dpm = (zf > 0) ? dpv[zf - 1] : 0.f;
    float m = Du - L * cpm;
    cp[zf] = Uu / m;
    dpv[zf] = (bb - L * dpm) / m;
    if (zf + 1 < NZW) __builtin_prefetch(&Cc[IC(zf + 2, y, x)], 0, 1);
  }
  float ztop = prof[145], zd = 0.75f * ztop;
  float xn = 0.f;
#pragma unroll
  for (int zf = NZW - 1; zf >= 0; zf--) {
    float xv = dpv[zf] - cp[zf] * xn;
    float zc = (prof[96 + 1 + zf] + phi1in[IW(zf, y, x)]) / cG;
    float dmp = 0.f;
    if (zc > zd) { float s = sinf(1.5707963f * (1.f - (ztop - zc) / (ztop - zd))); dmp = 0.2f * s * s; }
    W2n[IW(zf, y, x)] = xv / (1.f + dmp * dtau);
    xn = xv;
  }
}

// phi corrector: ph2n += (dtau*G/2/bzmu) * W2n
__global__ void kphif(const float* mu1in, const float* W2n, float dtau, float* ph2n) {
  long i = (long)blockIdx.x * blockDim.x + threadIdx.x;
  if (i >= SZW) return;
  int x = (int)(i % NXv), y = (int)((i / NXv) % NYv), zf = (int)(i / ((long)NXv * NYv));
  float bzmu = 0.5f * (muS(mu1in, zf, y, x) + muS(mu1in, zf + 1, y, x));
  ph2n[i] += dtau * cG * 0.5f / bzmu * W2n[i];
}

// ---------------- combine: out = input + carry ----------------
__global__ void kcomb(const float* Uin, const float* Vin, const float* Win, const float* Oin,
                      const float* Tin, const float* Pin, const float* Min,
                      const float* U2, const float* V2, const float* W2, const float* O2,
                      const float* T2, const float* P2, const float* pi2, float* out) {
  long i = (long)blockIdx.x * blockDim.x + threadIdx.x;
  long o = 0;
  if (i < SZU) { out[i] = Uin[i] + U2[i]; return; } i -= SZU; o += SZU;
  if (i < SZV) { out[o + i] = Vin[i] + V2[i]; return; } i -= SZV; o += SZV;
  if (i < SZW) { out[o + i] = Win[i] + W2[i]; return; } i -= SZW; o += SZW;
  if (i < SZW) { out[o + i] = Oin[i] + O2[i]; return; } i -= SZW; o += SZW;
  if (i < SZC) { out[o + i] = Tin[i] + T2[i]; return; } i -= SZC; o += SZC;
  if (i < SZW) { out[o + i] = Pin[i] + P2[i]; return; } i -= SZW; o += SZW;
  if (i < SZC) { long p = i % ((long)NXv * NYv); out[o + i] = Min[i] + pi2[p]; }
}

// ---------------- host orchestration ----------------
extern "C" void kernel_launch(void* const* d_in, const int* in_sizes, int n_in,
                              void* d_out, int out_size, void* d_ws, size_t ws_size,
                              hipStream_t stream) {
  (void)in_sizes; (void)n_in; (void)out_size; (void)ws_size;
  const float* Uin = (const float*)d_in[0];
  const float* Vin = (const float*)d_in[1];
  const float* Win = (const float*)d_in[2];
  const float* Oin = (const float*)d_in[3];
  const float* Tin = (const float*)d_in[4];
  const float* Phin = (const float*)d_in[5];
  const float* Min = (const float*)d_in[6];
  float* out = (float*)d_out;

  float* w = (float*)d_ws;
  size_t off = 0;
  auto alloc = [&](long n) { float* p = w + off; off += (size_t)n; return p; };
  float* prof = alloc(256);
  float* ALF = alloc(SZC);  float* PF = alloc(SZC);   float* CcF = alloc(SZC);
  float* GPTF = alloc(SZC); float* THF = alloc(SZC);  float* DZPHI = alloc(SZW);
  float* DPX = alloc(SZU);  float* DPY = alloc(SZV);
  float* ALR = alloc(SZC);  float* P1R = alloc(SZC);  float* THR = alloc(SZC);
  float* RU = alloc(SZU);   float* RV = alloc(SZV);   float* RW = alloc(SZW);
  float* RTh = alloc(SZC);  float* Rphi = alloc(SZW); float* RMu = alloc(SZC);
  float* U2b[2] = {alloc(SZU), alloc(SZU)};
  float* V2b[2] = {alloc(SZV), alloc(SZV)};
  float* W2b[2] = {alloc(SZW), alloc(SZW)};
  float* T2b[2] = {alloc(SZC), alloc(SZC)};
  float* P2b[2] = {alloc(SZW), alloc(SZW)};
  float* pi2b[2] = {alloc(SZP), alloc(SZP)};
  float* O2 = alloc(SZW);
  float* P2t = alloc(SZC);
  float* Dv = alloc(SZC);
  float* dpi = alloc(SZP);

  const int B = 256;
  dim3 gC((unsigned)((SZC + B - 1) / B)), gU((unsigned)((SZU + B - 1) / B));
  dim3 gV((unsigned)((SZV + B - 1) / B)), gW((unsigned)((SZW + B - 1) / B));
  dim3 gP((unsigned)((SZP + B - 1) / B));
  long TOT = SZU + SZV + 3 * SZW + 2 * SZC;
  dim3 gT((unsigned)((TOT + B - 1) / B));

  // output-field views (intermediate stage states also live in d_out)
  const float* sU = out;
  const float* sV = sU + SZU;
  const float* sW = sV + SZV;
  const float* sO = sW + SZW;
  const float* sT = sO + SZW;
  const float* sP = sT + SZC;
  const float* sM = sP + SZW;

  kprof<<<1, 64, 0, stream>>>(prof);
  kfixed<<<gC, B, 0, stream>>>(Tin, Phin, Min, prof, ALF, PF, CcF, GPTF, THF, DZPHI);
  kdpx<<<gU, B, 0, stream>>>(Phin, DPX);
  kdpy<<<gV, B, 0, stream>>>(Phin, DPY);

  auto rhs = [&](const float* aU, const float* aV, const float* aW, const float* aO,
                 const float* aT, const float* aP, const float* aM) {
    kderive<<<gC, B, 0, stream>>>(aT, aP, aM, prof, ALR, P1R, THR);
    krU<<<gU, B, 0, stream>>>(aU, aV, aO, aP, aM, ALR, P1R, RU);
    krV<<<gV, B, 0, stream>>>(aU, aV, aO, aP, aM, ALR, P1R, RV);
    krW<<<gW, B, 0, stream>>>(aU, aV, aW, aO, aM, P1R, RW);
    krTh<<<gC, B, 0, stream>>>(aU, aV, aO, THR, RTh);
    krPhi<<<gW, B, 0, stream>>>(aU, aV, aW, aO, aP, aM, prof, Rphi);
    krMu<<<gC, B, 0, stream>>>(aU, aV, aO, RMu);
  };

  auto stage = [&](float dtau, int ntau) {
    int cur = 0;
    hipMemsetAsync(U2b[0], 0, SZU * sizeof(float), stream);
    hipMemsetAsync(V2b[0], 0, SZV * sizeof(float), stream);
    hipMemsetAsync(W2b[0], 0, SZW * sizeof(float), stream);
    hipMemsetAsync(T2b[0], 0, SZC * sizeof(float), stream);
    hipMemsetAsync(P2b[0], 0, SZW * sizeof(float), stream);
    hipMemsetAsync(pi2b[0], 0, SZP * sizeof(float), stream);
    for (int it = 0; it < ntau; ++it) {
      int nx = cur ^ 1;
      kp2<<<gC, B, 0, stream>>>(T2b[cur], P2b[cur], pi2b[cur], Tin, Min, ALF, PF, P2t);
      kU2<<<gU, B, 0, stream>>>(U2b[cur], RU, P2t, P2b[cur], pi2b[cur], Min, ALF, DPX, dtau, U2b[nx]);
      kV2<<<gV, B, 0, stream>>>(V2b[cur], RV, P2t, P2b[cur], pi2b[cur], Min, ALF, DPY, dtau, V2b[nx]);
      kdpi<<<gP, B, 0, stream>>>(U2b[nx], V2b[nx], Uin, Vin, pi2b[cur], dtau, dpi, pi2b[nx]);
      kdv<<<gC, B, 0, stream>>>(U2b[nx], V2b[nx], RMu, Dv);
      kcum<<<dim3(NXv / 64, NYv), 128, 0, stream>>>(Dv, dpi, O2);
      kth2<<<gC, B, 0, stream>>>(T2b[cur], RTh, U2b[nx], V2b[nx], O2, THF, dtau, T2b[nx]);
      kpb<<<gW, B, 0, stream>>>(P2b[cur], Rphi, O2, DZPHI, W2b[cur], Min, dtau, P2b[nx]);
      kwsolve<<<gP, B, 0, stream>>>(CcF, GPTF, T2b[nx], T2b[cur], P2b[nx], P2b[cur], RW, W2b[cur],
                                    pi2b[nx], pi2b[cur], Min, Phin, prof, dtau, W2b[nx]);
      kphif<<<gW, B, 0, stream>>>(Min, W2b[nx], dtau, P2b[nx]);
      cur = nx;
    }
    kcomb<<<gT, B, 0, stream>>>(Uin, Vin, Win, Oin, Tin, Phin, Min,
                                U2b[cur], V2b[cur], W2b[cur], O2, T2b[cur], P2b[cur], pi2b[cur],
                                out);
  };

  // stage 1: R = RHS(inputs); s1 = step(DT/3, 1)
  rhs(Uin, Vin, Win, Oin, Tin, Phin, Min);
  stage(0.5f / 3.f, 1);
  // stage 2: R = RHS(s1); s2 = step(DT/4, 2)
  rhs(sU, sV, sW, sO, sT, sP, sM);
  stage(0.125f, 2);
  // stage 3: R = RHS(s2); out = step(DT/4, 4)
  rhs(sU, sV, sW, sO, sT, sP, sM);
  stage(0.125f, 4);
}